// GCN_Geo_1889785610772
// MI455X (gfx1250) — compile-verified
//
#include <hip/hip_runtime.h>
#include <hip/hip_bf16.h>

typedef float v2f __attribute__((ext_vector_type(2)));
typedef float v8f __attribute__((ext_vector_type(8)));

#define NV    32768      // B*N nodes
#define ETOT  32704      // B*(N-1) edges
#define NRES  4096       // B*A residues
#define K1PAD 588        // 45*13 = 585 padded to mult of 4
#define K2PAD 832        // 64*13

__device__ __forceinline__ v8f wmma_step(v8f c, float a0, float a1, float b0, float b1) {
  v2f a; a.x = a0; a.y = a1;
  v2f b; b.x = b0; b.y = b1;
  return __builtin_amdgcn_wmma_f32_16x16x4_f32(false, a, false, b, (short)0, c, false, false);
}

// ---------------- prep: build concatenated weights, init edge map, init xa ----------------
__global__ void prep_kernel(const float* __restrict__ root1, const float* __restrict__ linb1,
                            const float* __restrict__ linw1, const float* __restrict__ root2,
                            const float* __restrict__ linb2, const float* __restrict__ linw2,
                            const float* __restrict__ init_w, const float* __restrict__ arma_w,
                            const float* __restrict__ arma_root, const float* __restrict__ aa,
                            float* __restrict__ W1, float* __restrict__ W2,
                            float* __restrict__ WCAT, float* __restrict__ xa,
                            int* __restrict__ in_edge) {
  int idx = blockIdx.x * blockDim.x + threadIdx.x;
  const int SZ1 = K1PAD * 64;       // 37632
  const int SZ2 = K2PAD * 32;       // 26624
  const int SZC = 30 * 256 * 128;   // 983040
  const int SZI = NV;               // 32768
  const int SZX = NRES * 128;       // 524288
  if (idx < SZ1) {
    int r = idx / 64, o = idx % 64;
    float v;
    if (r < 45)       v = root1[r * 64 + o];
    else if (r < 90)  v = linb1[(r - 45) * 64 + o];
    else if (r < 585) { int z = r - 90; int d = z / 45, i = z % 45; v = linw1[d * 2880 + i * 64 + o]; }
    else              v = 0.0f;
    W1[idx] = v; return;
  }
  idx -= SZ1;
  if (idx < SZ2) {
    int r = idx / 32, o = idx % 32;
    float v;
    if (r < 64)       v = root2[r * 32 + o];
    else if (r < 128) v = linb2[(r - 64) * 32 + o];
    else              { int z = r - 128; int d = z / 64, i = z % 64; v = linw2[d * 2048 + i * 32 + o]; }
    W2[idx] = v; return;
  }
  idx -= SZ2;
  if (idx < SZC) {
    int w = idx / 32768, rem = idx % 32768;
    int r = rem / 128, o = rem % 128;
    int t = w / 3, k = w % 3;
    float v;
    if (r < 128) {
      if (t == 0) v = (r < 127) ? init_w[(k * 127 + r) * 128 + o] : 0.0f;
      else        v = arma_w[(((t - 1) * 3 + k) * 128 + r) * 128 + o];
    } else if (r < 255) {
      v = arma_root[((t * 3 + k) * 127 + (r - 128)) * 128 + o];
    } else v = 0.0f;
    WCAT[idx] = v; return;
  }
  idx -= SZC;
  if (idx < SZI) { in_edge[idx] = -1; return; }
  idx -= SZI;
  if (idx < SZX) {
    int q = idx / 128, c = idx % 128;
    float v = 0.0f;
    if (c >= 32 && c < 127) v = aa[q * 95 + (c - 32)];
    xa[idx] = v; return;
  }
}

__global__ void edge_map_kernel(const int* __restrict__ eidx, int* __restrict__ in_edge) {
  int e = blockIdx.x * blockDim.x + threadIdx.x;
  if (e < ETOT) in_edge[eidx[ETOT + e]] = e;   // dst row of edge_index; dsts are unique
}

// ---------------- NNConv layer 1: h1[NV,64] ----------------
__global__ __launch_bounds__(128) void layer1_kernel(
    const float* __restrict__ x, const float* __restrict__ ea,
    const int* __restrict__ eidx, const int* __restrict__ in_edge,
    const float* __restrict__ W, const float* __restrict__ bias,
    float* __restrict__ h1) {
  __shared__ float Alds[16 * K1PAD];
  const int v0 = blockIdx.x * 16;
  {
    const int m = threadIdx.x >> 3, sub = threadIdx.x & 7;
    const int v = v0 + m;
    const int e = in_edge[v];
    const float* xv = x + v * 45;
    const float* xs = xv;             // dummy; masked below
    float eav[11] = {};
    if (e >= 0) {
      xs = x + eidx[e] * 45;          // src row of edge_index
      #pragma unroll
      for (int d = 0; d < 11; ++d) eav[d] = ea[e * 11 + d];
    }
    float* row = Alds + m * K1PAD;
    for (int i = sub; i < 45; i += 8) {
      const float xvi = xv[i];
      const float xsi = (e >= 0) ? xs[i] : 0.0f;
      row[i] = xvi;
      row[45 + i] = xsi;
      #pragma unroll
      for (int d = 0; d < 11; ++d) row[90 + d * 45 + i] = eav[d] * xsi;
    }
    for (int i = 585 + sub; i < K1PAD; i += 8) row[i] = 0.0f;
  }
  __syncthreads();
  const int wave = threadIdx.x >> 5, lane = threadIdx.x & 31;
  const int half = lane >> 4, ln = lane & 15;
  const int n0 = wave * 16;
  const float* arow = Alds + ln * K1PAD + half * 2;
  v8f c = {};
  for (int s = 0; s < K1PAD / 4; ++s) {
    const int kr = s * 4 + half * 2;
    int pr = kr + 16; if (pr > K1PAD - 4) pr = K1PAD - 4;
    __builtin_prefetch(&W[pr * 64 + n0 + ln], 0, 1);
    c = wmma_step(c, arow[s * 4], arow[s * 4 + 1],
                  W[kr * 64 + n0 + ln], W[(kr + 1) * 64 + n0 + ln]);
  }
  const float bo = bias[n0 + ln];
  #pragma unroll
  for (int r = 0; r < 8; ++r) {
    const int mrow = r + half * 8;
    h1[(v0 + mrow) * 64 + n0 + ln] = fmaxf(c[r] + bo, 0.0f);
  }
}

// ---------------- NNConv layer 2: h2[NV,32] ----------------
__global__ __launch_bounds__(64) void layer2_kernel(
    const float* __restrict__ h1, const float* __restrict__ ea,
    const int* __restrict__ eidx, const int* __restrict__ in_edge,
    const float* __restrict__ W, const float* __restrict__ bias,
    float* __restrict__ h2) {
  __shared__ float Alds[16 * K2PAD];
  const int v0 = blockIdx.x * 16;
  {
    const int m = threadIdx.x >> 2, sub = threadIdx.x & 3;
    const int v = v0 + m;
    const int e = in_edge[v];
    const float* hv = h1 + v * 64;
    const float* hs = hv;
    float eav[11] = {};
    if (e >= 0) {
      hs = h1 + eidx[e] * 64;
      #pragma unroll
      for (int d = 0; d < 11; ++d) eav[d] = ea[e * 11 + d];
    }
    float* row = Alds + m * K2PAD;
    for (int i = sub; i < 64; i += 4) {
      const float hvi = hv[i];
      const float hsi = (e >= 0) ? hs[i] : 0.0f;
      row[i] = hvi;
      row[64 + i] = hsi;
      #pragma unroll
      for (int d = 0; d < 11; ++d) row[128 + d * 64 + i] = eav[d] * hsi;
    }
  }
  __syncthreads();
  const int wave = threadIdx.x >> 5, lane = threadIdx.x & 31;
  const int half = lane >> 4, ln = lane & 15;
  const int n0 = wave * 16;
  const float* arow = Alds + ln * K2PAD + half * 2;
  v8f c = {};
  for (int s = 0; s < K2PAD / 4; ++s) {
    const int kr = s * 4 + half * 2;
    int pr = kr + 16; if (pr > K2PAD - 4) pr = K2PAD - 4;
    __builtin_prefetch(&W[pr * 32 + n0 + ln], 0, 1);
    c = wmma_step(c, arow[s * 4], arow[s * 4 + 1],
                  W[kr * 32 + n0 + ln], W[(kr + 1) * 32 + n0 + ln]);
  }
  const float bo = bias[n0 + ln];
  #pragma unroll
  for (int r = 0; r < 8; ++r) {
    const int mrow = r + half * 8;
    h2[(v0 + mrow) * 32 + n0 + ln] = fmaxf(c[r] + bo, 0.0f);
  }
}

// ---------------- pool atoms -> residues (scatter using provided index arrays) ----------------
__global__ void pool_kernel(const float* __restrict__ h2, const int* __restrict__ idxb,
                            const int* __restrict__ mono, float* __restrict__ xa) {
  int idx = blockIdx.x * blockDim.x + threadIdx.x;
  if (idx >= NV * 32) return;
  int v = idx >> 5, c = idx & 31;
  int q = idxb[v] * 64 + mono[v];
  atomicAdd(&xa[q * 128 + c], h2[v * 32 + c]);
}

// ---- one ARMA iteration, 32 rows/block (2 M-tiles, 2 accumulators/wave, B reused 2x) ----
__global__ __launch_bounds__(256) void arma_kernel(
    const float* __restrict__ prevbuf, int prev_is_xa,
    const float* __restrict__ xa,
    const float* __restrict__ Wcat_t, const float* __restrict__ bias_t,
    float* __restrict__ dst) {
  __shared__ float Alds[32 * 256];
  const int k = blockIdx.x >> 7;        // 0..2
  const int tile = blockIdx.x & 127;    // 0..127
  const int q0 = tile * 32;
  const float* prev = prev_is_xa ? xa : (prevbuf + (size_t)k * NRES * 128);
  {
    const int m = threadIdx.x >> 3, sub = threadIdx.x & 7;   // 8 threads per row, 32 rows
    const int q = q0 + m, j = q & 63;
    float* row = Alds + m * 256;
    for (int i = sub; i < 128; i += 8) {
      row[i]       = (j >= 2) ? prev[(q - 1) * 128 + i] : 0.0f;  // norm=0 for j<2
      row[128 + i] = xa[q * 128 + i];
    }
  }
  __syncthreads();
  const int wave = threadIdx.x >> 5, lane = threadIdx.x & 31;
  const int half = lane >> 4, ln = lane & 15;
  const int n0 = wave * 16;
  const float* W = Wcat_t + k * 256 * 128;
  const float* arow0 = Alds + ln * 256 + half * 2;
  const float* arow1 = arow0 + 16 * 256;
  v8f c0 = {}, c1 = {};
  for (int s = 0; s < 64; ++s) {
    const int kr = s * 4 + half * 2;
    const float b0 = W[kr * 128 + n0 + ln];
    const float b1 = W[(kr + 1) * 128 + n0 + ln];
    int pr = kr + 16; if (pr > 252) pr = 252;
    __builtin_prefetch(&W[pr * 128 + n0 + ln], 0, 1);
    c0 = wmma_step(c0, arow0[s * 4], arow0[s * 4 + 1], b0, b1);
    c1 = wmma_step(c1, arow1[s * 4], arow1[s * 4 + 1], b0, b1);
  }
  const float bo = bias_t[k * 128 + n0 + ln];
  #pragma unroll
  for (int r = 0; r < 8; ++r) {
    const int mrow = r + half * 8;
    dst[((size_t)k * NRES + q0 + mrow) * 128 + n0 + ln]      = fmaxf(c0[r] + bo, 0.0f);
    dst[((size_t)k * NRES + q0 + 16 + mrow) * 128 + n0 + ln] = fmaxf(c1[r] + bo, 0.0f);
  }
}

// ---------------- readout: p[b,c] = sum_j mean_k out ----------------
__global__ void reduce_kernel(const float* __restrict__ outf, float* __restrict__ p) {
  int b = blockIdx.x, c = threadIdx.x;  // 64 blocks x 128 threads
  float acc = 0.0f;
  for (int k = 0; k < 3; ++k)
    for (int j = 0; j < 64; ++j)
      acc += outf[((size_t)k * NRES + b * 64 + j) * 128 + c];
  p[b * 128 + c] = acc * (1.0f / 3.0f);
}

// ---------------- FC head (tiny) ----------------
__global__ __launch_bounds__(256) void fc_kernel(
    const float* __restrict__ p,
    const float* __restrict__ w1, const float* __restrict__ b1,
    const float* __restrict__ w2, const float* __restrict__ b2,
    const float* __restrict__ w3, const float* __restrict__ b3,
    const float* __restrict__ w4, const float* __restrict__ b4,
    float* __restrict__ out) {
  __shared__ float buf0[256];
  __shared__ float buf1[256];
  const int t = threadIdx.x;
  for (int b = 0; b < 64; ++b) {
    if (t < 128) buf0[t] = p[b * 128 + t];
    __syncthreads();
    float a1 = b1[t];
    for (int i = 0; i < 128; ++i) a1 += buf0[i] * w1[i * 256 + t];
    buf1[t] = fmaxf(a1, 0.0f);
    __syncthreads();
    if (t < 128) {
      float a2 = b2[t];
      for (int i = 0; i < 256; ++i) a2 += buf1[i] * w2[i * 128 + t];
      buf0[t] = fmaxf(a2, 0.0f);
    }
    __syncthreads();
    if (t < 64) {
      float a3 = b3[t];
      for (int i = 0; i < 128; ++i) a3 += buf0[i] * w3[i * 64 + t];
      buf1[t] = fmaxf(a3, 0.0f);
    }
    __syncthreads();
    if (t == 0) {
      float a4 = b4[0];
      for (int i = 0; i < 64; ++i) a4 += buf1[i] * w4[i];
      out[b] = a4;
    }
    __syncthreads();
  }
}

extern "C" void kernel_launch(void* const* d_in, const int* in_sizes, int n_in,
                              void* d_out, int out_size, void* d_ws, size_t ws_size,
                              hipStream_t stream) {
  (void)in_sizes; (void)n_in; (void)out_size; (void)ws_size;
  const float* x         = (const float*)d_in[0];
  const float* ea        = (const float*)d_in[1];
  const float* aa        = (const float*)d_in[2];
  const float* nn1_lin_w = (const float*)d_in[3];
  const float* nn1_lin_b = (const float*)d_in[4];
  const float* nn1_root  = (const float*)d_in[5];
  const float* nn1_bias  = (const float*)d_in[6];
  const float* nn2_lin_w = (const float*)d_in[7];
  const float* nn2_lin_b = (const float*)d_in[8];
  const float* nn2_root  = (const float*)d_in[9];
  const float* nn2_bias  = (const float*)d_in[10];
  const float* arma_init = (const float*)d_in[11];
  const float* arma_w    = (const float*)d_in[12];
  const float* arma_root = (const float*)d_in[13];
  const float* arma_bias = (const float*)d_in[14];
  const float* fc1_w = (const float*)d_in[15]; const float* fc1_b = (const float*)d_in[16];
  const float* fc2_w = (const float*)d_in[17]; const float* fc2_b = (const float*)d_in[18];
  const float* fc3_w = (const float*)d_in[19]; const float* fc3_b = (const float*)d_in[20];
  const float* fc4_w = (const float*)d_in[21]; const float* fc4_b = (const float*)d_in[22];
  const int* eidx = (const int*)d_in[23];
  const int* idxb = (const int*)d_in[24];
  const int* mono = (const int*)d_in[25];

  float* ws   = (float*)d_ws;
  float* h1   = ws;
  float* h2   = h1   + (size_t)NV * 64;
  float* xa   = h2   + (size_t)NV * 32;
  float* outA = xa   + (size_t)NRES * 128;
  float* outB = outA + (size_t)3 * NRES * 128;
  float* W1   = outB + (size_t)3 * NRES * 128;
  float* W2   = W1   + (size_t)K1PAD * 64;
  float* WCAT = W2   + (size_t)K2PAD * 32;
  float* P    = WCAT + (size_t)30 * 256 * 128;
  int* in_edge = (int*)(P + 64 * 128);

  prep_kernel<<<6267, 256, 0, stream>>>(nn1_root, nn1_lin_b, nn1_lin_w,
      nn2_root, nn2_lin_b, nn2_lin_w, arma_init, arma_w, arma_root, aa,
      W1, W2, WCAT, xa, in_edge);
  edge_map_kernel<<<(ETOT + 255) / 256, 256, 0, stream>>>(eidx, in_edge);
  layer1_kernel<<<NV / 16, 128, 0, stream>>>(x, ea, eidx, in_edge, W1, nn1_bias, h1);
  layer2_kernel<<<NV / 16, 64, 0, stream>>>(h1, ea, eidx, in_edge, W2, nn2_bias, h2);
  pool_kernel<<<(NV * 32) / 256, 256, 0, stream>>>(h2, idxb, mono, xa);

  float* bufs[2] = {outA, outB};
  for (int t = 0; t < 10; ++t) {
    const float* prev = (t == 0) ? xa : bufs[(t - 1) & 1];
    arma_kernel<<<3 * 128, 256, 0, stream>>>(prev, (t == 0) ? 1 : 0, xa,
        WCAT + (size_t)t * 3 * 256 * 128, arma_bias + t * 3 * 128, bufs[t & 1]);
  }
  reduce_kernel<<<64, 128, 0, stream>>>(bufs[1], P);
  fc_kernel<<<1, 256, 0, stream>>>(P, fc1_w, fc1_b, fc2_w, fc2_b,
                                   fc3_w, fc3_b, fc4_w, fc4_b, (float*)d_out);
}